// SkillExecutionAdapter_8581344657491
// MI455X (gfx1250) — compile-verified
//
#include <hip/hip_runtime.h>

#define D_MODEL   2048
#define SKILL_DIM 128
#define B_BATCH   2
#define T_SEQ     2048

typedef __attribute__((ext_vector_type(2))) float v2f;
typedef __attribute__((ext_vector_type(4))) float v4f;
typedef __attribute__((ext_vector_type(8))) float v8f;

// ---------------------------------------------------------------------------
// Kernel A: v[b][j] = b_kv[D + j] + sum_k skill[b][k] * W_kv[k][D + j]
// (only the V half of the kv projection matters; K is provably irrelevant)
// 4096 outputs, each a length-128 dot. Coalesced over j.
// ---------------------------------------------------------------------------
__global__ void skill_v_kernel(const float* __restrict__ skill,
                               const float* __restrict__ W_kv,
                               const float* __restrict__ b_kv,
                               float* __restrict__ v) {
    int idx = blockIdx.x * blockDim.x + threadIdx.x;   // 0 .. 4095
    int b = idx >> 11;                                  // / 2048
    int j = idx & (D_MODEL - 1);
    float acc = b_kv[D_MODEL + j];
    const float* srow = skill + b * SKILL_DIM;
    const float* wcol = W_kv + D_MODEL + j;             // column j of V-half
#pragma unroll 8
    for (int k = 0; k < SKILL_DIM; ++k) {
        acc = fmaf(srow[k], wcol[(size_t)k * (2 * D_MODEL)], acc);
    }
    v[idx] = acc;
}

// ---------------------------------------------------------------------------
// Kernel B: y = v_pad(16 x 2048) @ W_out(2048 x 2048), fp32 exact, via
// V_WMMA_F32_16X16X4_F32. One wave (32 lanes) per 16-column output tile.
//
// A (16x4 f32) layout (ISA 7.12.2): lanes 0-15 hold {K=0,K=1}, lanes 16-31
// hold {K=2,K=3} for row M = lane&15. Rows M>=2 are zero padding.
// B (4x16) mirrored: lane n holds {K=kb, K=kb+1} of column N = lane&15.
// D (16x16 f32): VGPR r, lanes 0-15 -> M=r; we need M=0 (acc[0]) and
// M=1 (acc[1]) on lanes 0-15.
// Two accumulators break the WMMA->WMMA RAW dependency chain.
// ---------------------------------------------------------------------------
__global__ void __launch_bounds__(32)
gemv_wmma_kernel(const float* __restrict__ v,
                 const float* __restrict__ W_out,
                 float* __restrict__ y) {
    const int lane  = threadIdx.x;            // 0..31
    const int j0    = blockIdx.x * 16;        // output-column tile
    const int m     = lane & 15;              // row (A) / col (B) within tile
    const int khalf = (lane >> 4) << 1;       // 0 for lanes 0-15, 2 for 16-31
    const bool arow = (m < B_BATCH);
    const float* vrow = v + m * D_MODEL;      // only read when arow

    v8f acc0 = {};
    v8f acc1 = {};

#pragma unroll 2
    for (int k = 0; k < D_MODEL; k += 8) {
        // prefetch the W_out stream a few lines ahead (global_prefetch)
        if (k + 16 < D_MODEL) {
            __builtin_prefetch(&W_out[(size_t)(k + 16 + khalf) * D_MODEL + j0 + m], 0, 1);
        }
        {
            const int kb = k + khalf;
            v2f a, bm;
            a.x  = arow ? vrow[kb]     : 0.0f;
            a.y  = arow ? vrow[kb + 1] : 0.0f;
            bm.x = W_out[(size_t)kb       * D_MODEL + j0 + m];
            bm.y = W_out[(size_t)(kb + 1) * D_MODEL + j0 + m];
            acc0 = __builtin_amdgcn_wmma_f32_16x16x4_f32(
                false, a, false, bm, (short)0, acc0, false, false);
        }
        {
            const int kb = k + 4 + khalf;
            v2f a, bm;
            a.x  = arow ? vrow[kb]     : 0.0f;
            a.y  = arow ? vrow[kb + 1] : 0.0f;
            bm.x = W_out[(size_t)kb       * D_MODEL + j0 + m];
            bm.y = W_out[(size_t)(kb + 1) * D_MODEL + j0 + m];
            acc1 = __builtin_amdgcn_wmma_f32_16x16x4_f32(
                false, a, false, bm, (short)0, acc1, false, false);
        }
    }

    v8f acc = acc0 + acc1;
    if (lane < 16) {
        y[0 * D_MODEL + j0 + lane] = acc[0];   // M=0 row -> batch 0
        y[1 * D_MODEL + j0 + lane] = acc[1];   // M=1 row -> batch 1
    }
}

// ---------------------------------------------------------------------------
// Kernel C: out[b, t, :] = y[b, :]  — 32 MiB of pure-bandwidth broadcast.
// float4 non-temporal stores; y (16 KiB) stays resident in cache.
// Total (B*T*D)/4 = 2^21 float4 elements; per-batch T*D/4 = 2^20.
// ---------------------------------------------------------------------------
__global__ void broadcast_kernel(const v4f* __restrict__ y4,
                                 v4f* __restrict__ out4) {
    size_t i = (size_t)blockIdx.x * blockDim.x + threadIdx.x;  // 0 .. 2^21-1
    int b  = (int)(i >> 20);                                   // batch
    int j4 = (int)(i & 511);                                   // D/4 = 512
    v4f val = y4[(size_t)b * 512 + j4];
    __builtin_nontemporal_store(val, &out4[i]);
}

// ---------------------------------------------------------------------------
// inputs (setup_inputs order): 0=h, 1=skill, 2=W_kv, 3=b_kv, 4=W_q, 5=W_out
// h and W_q are mathematically irrelevant (softmax over identical keys is
// uniform; uniform average of identical values is the value itself).
// ---------------------------------------------------------------------------
extern "C" void kernel_launch(void* const* d_in, const int* in_sizes, int n_in,
                              void* d_out, int out_size, void* d_ws, size_t ws_size,
                              hipStream_t stream) {
    const float* skill = (const float*)d_in[1];
    const float* W_kv  = (const float*)d_in[2];
    const float* b_kv  = (const float*)d_in[3];
    const float* W_out = (const float*)d_in[5];

    float* ws = (float*)d_ws;
    float* v  = ws;                    // B*D      = 4096 floats
    float* y  = ws + B_BATCH * D_MODEL; // B*D     = 4096 floats

    // Step 1: v = skill @ W_kv[:, D:2D] + b_kv[D:2D]
    skill_v_kernel<<<(B_BATCH * D_MODEL) / 256, 256, 0, stream>>>(skill, W_kv, b_kv, v);

    // Step 2: y = v @ W_out  (fp32 WMMA, one wave per 16-col tile)
    gemv_wmma_kernel<<<D_MODEL / 16, 32, 0, stream>>>(v, W_out, y);

    // Step 3: broadcast y over the T axis into the output
    broadcast_kernel<<<(B_BATCH * T_SEQ * D_MODEL / 4) / 256, 256, 0, stream>>>(
        (const v4f*)y, (v4f*)d_out);
}